// GLNorm2_56427280335258
// MI455X (gfx1250) — compile-verified
//
#include <hip/hip_runtime.h>

typedef __attribute__((ext_vector_type(2))) float v2f;
typedef __attribute__((ext_vector_type(4))) float v4f;
typedef __attribute__((ext_vector_type(8))) float v8f;

#define NBATCH   128          // A*B = 8*16
#define NROWS    1024         // N = M
#define KDIM     64
#define ROWCHUNK 128          // rows reduced per partial-sum block
#define NCHUNK   (NROWS / ROWCHUNK)   // 8

// ---------------------------------------------------------------------------
// Pass 1a: partial column sums. grid = 128 batches * 2 tensors * 8 chunks.
// Each block reduces 128 rows x 64 cols of one tensor slice -> 64 partials.
// bid encodes batch*16 + tensor*8 + chunk, so partials land contiguously.
// ---------------------------------------------------------------------------
__global__ void GLNorm2_partial_sums(const float* __restrict__ u,
                                     const float* __restrict__ v,
                                     float* __restrict__ wsP) {
  int bid    = blockIdx.x;
  int chunk  = bid & 7;
  int tensor = (bid >> 3) & 1;
  int batch  = bid >> 4;
  const float* src = (tensor == 0 ? u : v) + (size_t)batch * NROWS * KDIM
                                           + (size_t)chunk * ROWCHUNK * KDIM;
  int tid  = threadIdx.x;
  int k    = tid & 63;        // column
  int part = tid >> 6;        // 0..3, row phase

  float acc = 0.f;
  for (int r = part; r < ROWCHUNK; r += 4)
    acc += src[(size_t)r * KDIM + k];

  __shared__ float red[256];
  red[tid] = acc;
  __syncthreads();
  if (part == 0) {
    float s = red[k] + red[k + 64] + red[k + 128] + red[k + 192];
    wsP[(size_t)bid * 64 + k] = s;
  }
}

// ---------------------------------------------------------------------------
// Pass 1b: fold the 8 chunk partials. 16384 output values (128 batches * 2
// tensors * 64 k). Deterministic (fixed order), no atomics.
// wsS layout per batch: [0:64] = u_sum, [64:128] = v_sum.
// ---------------------------------------------------------------------------
__global__ void GLNorm2_final_sums(const float* __restrict__ wsP,
                                   float* __restrict__ wsS) {
  int gid   = blockIdx.x * 256 + threadIdx.x;   // 0..16383
  int batch = gid >> 7;
  int j     = gid & 127;                        // tensor*64 + k
  int tensor = j >> 6;
  int k      = j & 63;
  const float* p = wsP + ((size_t)(batch * 16 + tensor * 8)) * 64 + k;
  float s = 0.f;
#pragma unroll
  for (int c = 0; c < 8; ++c) s += p[(size_t)c * 64];
  wsS[(size_t)batch * 128 + j] = s;
}

// ---------------------------------------------------------------------------
// Pass 2: per-row dot with the opposite tensor's column sum via
// V_WMMA_F32_16X16X4_F32, then gated streaming copy.
// grid = 128 batches * 2 sides * 8 row-slabs; block = 256 (8 waves).
// Each wave owns a 16-row tile: 16 accumulating WMMAs cover K=64.
//   A (16x4 f32): lane l holds row m=l&15; vgpr0 = K0+2*(l>>4), vgpr1 = +1
//                 -> one 8B load per lane per chunk, each element read once.
//   B (4x16 f32): sum vector replicated across all 16 columns; lower lanes
//                 hold lower K, mirroring A's layout.
//   D: every column equals row_sum; rows 0-7 in lanes 0-15 (vgpr r = row r),
//      rows 8-15 in lanes 16-31.
// ---------------------------------------------------------------------------
__global__ void GLNorm2_apply(const float* __restrict__ u,
                              const float* __restrict__ v,
                              const float* __restrict__ wsS,
                              float* __restrict__ out) {
  int bid    = blockIdx.x;
  int rowblk = bid & 7;          // 128-row slab within the slice
  int side   = (bid >> 3) & 1;   // 0: gate u rows, 1: gate v rows
  int batch  = bid >> 4;

  const size_t sliceElems = (size_t)NROWS * KDIM;
  const float* src  = (side == 0 ? u : v) + (size_t)batch * sliceElems
                                          + (size_t)rowblk * 128 * KDIM;
  // u rows dot v_sum; v rows dot u_sum
  const float* sums = wsS + (size_t)batch * 128 + (side == 0 ? 64 : 0);
  float* dst = out + (size_t)side * NBATCH * sliceElems
                   + (size_t)batch * sliceElems
                   + (size_t)rowblk * 128 * KDIM;

  int tid  = threadIdx.x;
  int lane = tid & 31;
  int wave = tid >> 5;           // 0..7, each wave = one 16-row tile
  int hi   = lane >> 4;          // lane half selects K offset +2
  int m    = lane & 15;          // matrix row within tile

  // Preload replicated B operands (uniform per lane-half; stays in VGPRs).
  v2f bvec[16];
#pragma unroll
  for (int c = 0; c < 16; ++c)
    bvec[c] = *(const v2f*)(sums + 4 * c + 2 * hi);

  const float* rowbase = src + (size_t)(wave * 16 + m) * KDIM;
  v8f acc = {};
#pragma unroll
  for (int c = 0; c < 16; ++c) {
    v2f a = *(const v2f*)(rowbase + 4 * c + 2 * hi);
    // 8 args: (neg_a, A, neg_b, B, c_mod, C, reuse_a, reuse_b)
    acc = __builtin_amdgcn_wmma_f32_16x16x4_f32(
        false, a, false, bvec[c], (short)0, acc, false, false);
  }

  // Broadcast the 16 per-row sums of each wave's tile through LDS.
  __shared__ float rs[8][16];
  if ((lane & 15) == 0) {
    int off = hi * 8;            // lane 0 -> rows 0-7, lane 16 -> rows 8-15
#pragma unroll
    for (int r = 0; r < 8; ++r) rs[wave][off + r] = acc[r];
  }
  __syncthreads();

  // Gated streaming copy: 128 rows * 64 cols = 2048 float4 slots, fully
  // coalesced (consecutive lanes -> consecutive 16B quads). Source re-read
  // hits L0/L2 (just loaded as WMMA A operands).
#pragma unroll
  for (int i = 0; i < 8; ++i) {
    int idx  = i * 256 + tid;    // 0..2047
    int row  = idx >> 4;         // 0..127
    int quad = idx & 15;
    float gate = rs[row >> 4][row & 15] > 0.0f ? 1.0f : 0.0f;
    v4f val = *(const v4f*)(src + (size_t)row * KDIM + quad * 4);
    val *= gate;
    *(v4f*)(dst + (size_t)row * KDIM + quad * 4) = val;
  }
}

// ---------------------------------------------------------------------------
extern "C" void kernel_launch(void* const* d_in, const int* in_sizes, int n_in,
                              void* d_out, int out_size, void* d_ws, size_t ws_size,
                              hipStream_t stream) {
  const float* u = (const float*)d_in[0];
  const float* v = (const float*)d_in[1];
  float* out = (float*)d_out;

  float* wsP = (float*)d_ws;            // 2048 * 64 floats   (512 KB)
  float* wsS = wsP + 2048 * 64;         // 128 * 128 floats   (64 KB)

  GLNorm2_partial_sums<<<NBATCH * 2 * NCHUNK, 256, 0, stream>>>(u, v, wsP);
  GLNorm2_final_sums<<<64, 256, 0, stream>>>(wsP, wsS);
  GLNorm2_apply<<<NBATCH * 2 * NCHUNK, 256, 0, stream>>>(u, v, wsS, out);
}